// MixedScore_MultiHeadAttention_10617159156310
// MI455X (gfx1250) — compile-verified
//
#include <hip/hip_runtime.h>

// CDNA5 / gfx1250: wave32, WMMA 16x16x4 f32 + packed-f32 (V_PK_FMA_F32) MLP,
// online softmax with deferred (per-lane partial) denominator.
typedef __attribute__((ext_vector_type(2))) float v2f;
typedef __attribute__((ext_vector_type(8))) float v8f;

#define B_ 4
#define H_ 16
#define R_ 512
#define C_ 512
#define D_ 16
#define M_ 16

__device__ __forceinline__ float group16_max(float x) {
  x = fmaxf(x, __shfl_xor(x, 1, 32));
  x = fmaxf(x, __shfl_xor(x, 2, 32));
  x = fmaxf(x, __shfl_xor(x, 4, 32));
  x = fmaxf(x, __shfl_xor(x, 8, 32));
  return x;
}
__device__ __forceinline__ float group16_sum(float x) {
  x += __shfl_xor(x, 1, 32);
  x += __shfl_xor(x, 2, 32);
  x += __shfl_xor(x, 4, 32);
  x += __shfl_xor(x, 8, 32);
  return x;
}

__global__ __launch_bounds__(256)
void mixed_attn_kernel(const float* __restrict__ q,
                       const float* __restrict__ k,
                       const float* __restrict__ v,
                       const float* __restrict__ cost,
                       const float* __restrict__ w1,
                       const float* __restrict__ b1,
                       const float* __restrict__ w2,
                       const float* __restrict__ b2,
                       float* __restrict__ out) {
  // per-wave 16x16 P-transpose staging, padded to 17 to avoid bank conflicts
  __shared__ float plds[8][16 * 17];

  const int wave = threadIdx.x >> 5;
  const int lane = threadIdx.x & 31;
  const int lr   = lane & 15;   // index within 16-lane group
  const int hl   = lane >> 4;   // lane half (0/1)

  const int blk = blockIdx.x;
  const int rb  = blk & 3;          // 128-row block within R
  const int h   = (blk >> 2) & 15;  // head
  const int b   = blk >> 6;         // batch

  const int row_base = rb * 128 + wave * 16;

  const float* qh    = q + (size_t)(b * H_ + h) * R_ * D_;
  const float* kh    = k + (size_t)(b * H_ + h) * C_ * D_;
  const float* vh    = v + (size_t)(b * H_ + h) * C_ * D_;
  const float* costb = cost + (size_t)b * R_ * C_;

  // per-head MLP parameters, packed in pairs for V_PK_FMA_F32 (all 8B aligned)
  v2f w1a2[M_ / 2], w1b2[M_ / 2], b12[M_ / 2], w22[M_ / 2];
#pragma unroll
  for (int mm = 0; mm < M_ / 2; ++mm) {
    w1a2[mm] = *reinterpret_cast<const v2f*>(w1 + h * 2 * M_ + 2 * mm);       // dot weights
    w1b2[mm] = *reinterpret_cast<const v2f*>(w1 + h * 2 * M_ + M_ + 2 * mm);  // cost weights
    b12[mm]  = *reinterpret_cast<const v2f*>(b1 + h * M_ + 2 * mm);
    w22[mm]  = *reinterpret_cast<const v2f*>(w2 + h * M_ + 2 * mm);
  }
  const float bb2 = b2[h];
  const v2f zero2 = {0.0f, 0.0f};

  // Q tile, A-layout for V_WMMA_F32_16X16X4_F32:
  // lane holds row (lr), VGPR v holds K = 4*kk + 2*hl + v  (even -> 8B aligned)
  v2f qa[4];
#pragma unroll
  for (int kk = 0; kk < 4; ++kk)
    qa[kk] = *reinterpret_cast<const v2f*>(qh + (row_base + lr) * D_ + 4 * kk + 2 * hl);

  v8f o = {};            // O accumulator, D-layout: r = w + 8*hl, d = lr
  float mrun[8], lrun[8];  // running row max / per-lane partial exp-sum
#pragma unroll
  for (int w = 0; w < 8; ++w) { mrun[w] = -3.0e38f; lrun[w] = 0.0f; }

  for (int c0 = 0; c0 < C_; c0 += 16) {
    // K tile, B-layout: lane holds column c0+lr, VGPR v holds K(=d) = 4*kk + 2*hl + v
    v2f kb[4];
#pragma unroll
    for (int kk = 0; kk < 4; ++kk)
      kb[kk] = *reinterpret_cast<const v2f*>(kh + (c0 + lr) * D_ + 4 * kk + 2 * hl);

    // S = Q * K^T  (K-dim 16 = 4 chained 16x16x4 f32 WMMAs)
    v8f s = {};
#pragma unroll
    for (int kk = 0; kk < 4; ++kk)
      s = __builtin_amdgcn_wmma_f32_16x16x4_f32(false, qa[kk], false, kb[kk],
                                                (short)0, s, false, false);

    // mixed score MLP (packed f32 pairs) + online softmax
    // D-layout: r = w + 8*hl, c = c0 + lr
    float p[8];
#pragma unroll
    for (int w = 0; w < 8; ++w) {
      const int r = row_base + w + 8 * hl;
      const float dot = s[w] * 0.25f;                       // 1/sqrt(D)
      const float cst = costb[(size_t)r * C_ + (c0 + lr)];
      const v2f dot2 = {dot, dot};
      const v2f cst2 = {cst, cst};
      v2f acc2 = {bb2, 0.0f};
#pragma unroll
      for (int mm = 0; mm < M_ / 2; ++mm) {
        v2f hv = __builtin_elementwise_fma(cst2, w1b2[mm], b12[mm]);
        hv = __builtin_elementwise_fma(dot2, w1a2[mm], hv);
        hv = __builtin_elementwise_max(hv, zero2);
        acc2 = __builtin_elementwise_fma(hv, w22[mm], acc2);
      }
      const float acc = acc2[0] + acc2[1];

      // row max must be group-uniform (rescale of o[w] is per-row);
      // denominator is kept as per-lane partial and reduced once at the end.
      const float tmax = group16_max(acc);
      const float mnew = fmaxf(mrun[w], tmax);
      const float resc = __expf(mrun[w] - mnew);
      const float pe   = __expf(acc - mnew);
      lrun[w] = fmaf(lrun[w], resc, pe);
      mrun[w] = mnew;
      o[w] *= resc;
      p[w] = pe;
    }

    // transpose P: D-layout -> A-layout via per-wave LDS tile
#pragma unroll
    for (int w = 0; w < 8; ++w)
      plds[wave][(w + 8 * hl) * 17 + lr] = p[w];
    __builtin_amdgcn_wave_barrier();
    asm volatile("s_wait_dscnt 0x0" ::: "memory");

    v2f pa[4], vb[4];
#pragma unroll
    for (int kk = 0; kk < 4; ++kk) {
      const int cl = 4 * kk + 2 * hl;
      pa[kk][0] = plds[wave][lr * 17 + cl];
      pa[kk][1] = plds[wave][lr * 17 + cl + 1];
      // V tile, B-layout: lane holds d = lr, K(=c local) = cl + v
      vb[kk][0] = vh[(c0 + cl) * D_ + lr];
      vb[kk][1] = vh[(c0 + cl + 1) * D_ + lr];
    }
    __builtin_amdgcn_wave_barrier();

    // O += P * V
#pragma unroll
    for (int kk = 0; kk < 4; ++kk)
      o = __builtin_amdgcn_wmma_f32_16x16x4_f32(false, pa[kk], false, vb[kk],
                                                (short)0, o, false, false);
  }

  // final denominator: one butterfly per row group, then write
  // out[b][r][h*16 + d] = O / l
#pragma unroll
  for (int w = 0; w < 8; ++w) {
    const float lsum = group16_sum(lrun[w]);
    const int r = row_base + w + 8 * hl;
    out[((size_t)b * R_ + r) * (H_ * D_) + h * D_ + lr] = o[w] / lsum;
  }
}

extern "C" void kernel_launch(void* const* d_in, const int* in_sizes, int n_in,
                              void* d_out, int out_size, void* d_ws, size_t ws_size,
                              hipStream_t stream) {
  const float* q    = (const float*)d_in[0];
  const float* k    = (const float*)d_in[1];
  const float* v    = (const float*)d_in[2];
  const float* cost = (const float*)d_in[3];
  const float* w1   = (const float*)d_in[4];
  const float* b1   = (const float*)d_in[5];
  const float* w2   = (const float*)d_in[6];
  const float* b2   = (const float*)d_in[7];
  float* out = (float*)d_out;

  // grid: B * H * (R/128) = 4 * 16 * 4 = 256 blocks; 256 threads = 8 wave32
  dim3 grid(B_ * H_ * (R_ / 128));
  dim3 block(256);
  mixed_attn_kernel<<<grid, block, 0, stream>>>(q, k, v, cost, w1, b1, w2, b2, out);
}